// ProteinGCNModel_29326036697585
// MI455X (gfx1250) — compile-verified
//
#include <hip/hip_runtime.h>

typedef __attribute__((ext_vector_type(2))) float v2f;
typedef __attribute__((ext_vector_type(8))) float v8f;

#define D 128  // feature dim (both layers)

// ---------------------------------------------------------------------------
// Degree / normalization kernels
// ---------------------------------------------------------------------------
__global__ __launch_bounds__(256) void k_deg_init(float* __restrict__ deg, int n) {
  int i = blockIdx.x * 256 + threadIdx.x;
  if (i < n) deg[i] = 1.0f;  // self-loop weight
}

__global__ __launch_bounds__(256) void k_deg_acc(const int* __restrict__ dst,
                                                 const float* __restrict__ ew,
                                                 float* __restrict__ deg, int e) {
  int i = blockIdx.x * 256 + threadIdx.x;
  if (i < e) atomicAdd(&deg[dst[i]], ew[i]);
}

__global__ __launch_bounds__(256) void k_deg_rsqrt(float* __restrict__ deg, int n) {
  int i = blockIdx.x * 256 + threadIdx.x;
  if (i < n) deg[i] = __frsqrt_rn(deg[i]);  // deg >= 1 always
}

__global__ __launch_bounds__(256) void k_zero(float* __restrict__ p, int n) {
  int i = blockIdx.x * 256 + threadIdx.x;
  if (i < n) p[i] = 0.0f;
}

// ---------------------------------------------------------------------------
// FP32 WMMA GEMM: H[N x 128] = X[N x 128] @ W[128 x 128]
// One wave computes one 16x16 tile. blockDim=256 (8 waves, wave32),
// wave w handles column tile w (8 * 16 = 128 columns). grid.x = N/16.
//
// V_WMMA_F32_16X16X4_F32 fragment layouts (ISA 7.12.2):
//   A (16x4, MxK):  lane = half*16 + m;  a[v] = A[m][half*2 + v], v in {0,1}
//   B (4x16, KxN):  lane = half*16 + n;  b[v] = B[half*2 + v][n]
//   C/D (16x16):    lanes 0-15 N=lane, lanes 16-31 N=lane-16;
//                   VGPR v -> M = v + half*8
// ---------------------------------------------------------------------------
__global__ __launch_bounds__(256) void k_gemm_wmma(const float* __restrict__ X,
                                                   const float* __restrict__ W,
                                                   float* __restrict__ H) {
  const int wave = threadIdx.x >> 5;
  const int lane = threadIdx.x & 31;
  const int m    = lane & 15;          // A row within tile / B-N / C-N
  const int half = lane >> 4;          // K-pair select
  const int row0 = blockIdx.x * 16;
  const int col0 = wave * 16;

  const float* __restrict__ xrow = X + (size_t)(row0 + m) * D;

  v8f acc = {};
#pragma unroll
  for (int k = 0; k < D; k += 4) {
    const int kk = k + half * 2;
    v2f a, b;
    a.x = xrow[kk];
    a.y = xrow[kk + 1];
    b.x = W[(size_t)kk * D + col0 + m];
    b.y = W[(size_t)(kk + 1) * D + col0 + m];
    acc = __builtin_amdgcn_wmma_f32_16x16x4_f32(
        /*neg_a=*/false, a, /*neg_b=*/false, b,
        /*c_mod=*/(short)0, acc, /*reuse_a=*/false, /*reuse_b=*/false);
  }

#pragma unroll
  for (int v = 0; v < 8; ++v) {
    const int r = row0 + v + half * 8;
    H[(size_t)r * D + col0 + m] = acc[v];
  }
}

// ---------------------------------------------------------------------------
// Edge scatter: one wave per edge; each lane handles 4 contiguous features.
//   agg[dst] += (dinv[src]*ew*dinv[dst]) * h[src]
// ---------------------------------------------------------------------------
__global__ __launch_bounds__(256) void k_scatter(const int* __restrict__ src,
                                                 const int* __restrict__ dst,
                                                 const float* __restrict__ ew,
                                                 const float* __restrict__ dinv,
                                                 const float* __restrict__ h,
                                                 float* __restrict__ agg, int e) {
  const int wid  = (blockIdx.x * 256 + threadIdx.x) >> 5;
  const int lane = threadIdx.x & 31;
  if (wid >= e) return;
  const int s = src[wid];
  const int d = dst[wid];
  const float nrm = dinv[s] * ew[wid] * dinv[d];
  const float4 v = ((const float4*)(h + (size_t)s * D))[lane];
  float* ad = agg + (size_t)d * D + lane * 4;
  atomicAdd(ad + 0, nrm * v.x);
  atomicAdd(ad + 1, nrm * v.y);
  atomicAdd(ad + 2, nrm * v.z);
  atomicAdd(ad + 3, nrm * v.w);
}

// ---------------------------------------------------------------------------
// Epilogue 1: h <- relu(agg + h*dinv^2 + b)   (in place over h)
// ---------------------------------------------------------------------------
__global__ __launch_bounds__(256) void k_epilogue_relu(const float* __restrict__ agg,
                                                       float* __restrict__ h,
                                                       const float* __restrict__ dinv,
                                                       const float* __restrict__ b,
                                                       int total) {
  int idx = blockIdx.x * 256 + threadIdx.x;
  if (idx >= total) return;
  const int i = idx >> 7;  // /128
  const int f = idx & (D - 1);
  const float di = dinv[i];
  const float v = agg[idx] + h[idx] * di * di + b[f];
  h[idx] = v > 0.0f ? v : 0.0f;
}

// ---------------------------------------------------------------------------
// Epilogue 2: out <- out + h2*dinv^2 + b      (out already holds scattered agg)
// ---------------------------------------------------------------------------
__global__ __launch_bounds__(256) void k_epilogue_out(float* __restrict__ out,
                                                      const float* __restrict__ h2,
                                                      const float* __restrict__ dinv,
                                                      const float* __restrict__ b,
                                                      int total) {
  int idx = blockIdx.x * 256 + threadIdx.x;
  if (idx >= total) return;
  const int i = idx >> 7;
  const int f = idx & (D - 1);
  const float di = dinv[i];
  out[idx] = out[idx] + h2[idx] * di * di + b[f];
}

// ---------------------------------------------------------------------------
extern "C" void kernel_launch(void* const* d_in, const int* in_sizes, int n_in,
                              void* d_out, int out_size, void* d_ws, size_t ws_size,
                              hipStream_t stream) {
  const float* x   = (const float*)d_in[0];
  const int*   ei  = (const int*)d_in[1];
  const float* ew  = (const float*)d_in[2];
  const float* W1  = (const float*)d_in[3];
  const float* b1  = (const float*)d_in[4];
  const float* W2  = (const float*)d_in[5];
  const float* b2  = (const float*)d_in[6];
  float*       out = (float*)d_out;

  const int N = in_sizes[0] / D;   // 50000 (multiple of 16)
  const int E = in_sizes[2];       // 600000
  const int total = N * D;

  const int* src = ei;
  const int* dst = ei + E;

  // Workspace layout
  float* dinv = (float*)d_ws;                  // N
  float* hbuf = dinv + N;                      // N*D
  float* aggb = hbuf + (size_t)N * D;          // N*D

  const int gN   = (N + 255) / 256;
  const int gE   = (E + 255) / 256;
  const int gEl  = (total + 255) / 256;
  const int gEw  = (E * 32 + 255) / 256;       // wave-per-edge
  const int gMM  = N / 16;

  // Normalization: deg -> dinv (stored in dinv buffer throughout)
  k_deg_init <<<gN,  256, 0, stream>>>(dinv, N);
  k_deg_acc  <<<gE,  256, 0, stream>>>(dst, ew, dinv, E);
  k_deg_rsqrt<<<gN,  256, 0, stream>>>(dinv, N);

  // ---- Layer 1 ----
  k_gemm_wmma<<<gMM, 256, 0, stream>>>(x, W1, hbuf);          // h1 = x @ W1
  k_zero     <<<gEl, 256, 0, stream>>>(aggb, total);
  k_scatter  <<<gEw, 256, 0, stream>>>(src, dst, ew, dinv, hbuf, aggb, E);
  k_epilogue_relu<<<gEl, 256, 0, stream>>>(aggb, hbuf, dinv, b1, total);
  // hbuf now holds relu(gcn1)

  // ---- Layer 2 ----
  k_gemm_wmma<<<gMM, 256, 0, stream>>>(hbuf, W2, aggb);       // h2 = h @ W2
  k_zero     <<<gEl, 256, 0, stream>>>(out, total);
  k_scatter  <<<gEw, 256, 0, stream>>>(src, dst, ew, dinv, aggb, out, E);
  k_epilogue_out<<<gEl, 256, 0, stream>>>(out, aggb, dinv, b2, total);
}